// Trainer_30013231464538
// MI455X (gfx1250) — compile-verified
//
#include <hip/hip_runtime.h>
#include <hip/hip_bf16.h>

typedef float v2f __attribute__((ext_vector_type(2)));
typedef float v8f __attribute__((ext_vector_type(8)));

#define LAM 0.05f
#define D_MIN 1.0f
#define EPS_D 1e-12f

// LDS staging geometry for the embedding-distance kernel
#define KC 64                 // K-slice of prototypes staged in LDS
#define ROWPITCH (KC + 2)     // +2 floats pad: row stride 66 banks -> conflict-free
#define NPROTO 128

__device__ __forceinline__ float wave_sum32(float v) {
    v += __shfl_xor(v, 16, 32);
    v += __shfl_xor(v, 8, 32);
    v += __shfl_xor(v, 4, 32);
    v += __shfl_xor(v, 2, 32);
    v += __shfl_xor(v, 1, 32);
    return v;
}

// ---------------------------------------------------------------------------
// ws layout (floats): [0]=ceSum [1]=mseSum [2]=divSum [3]=covSum
//                     [4..4+P)   colMin bits (uint view, init +inf)
// ---------------------------------------------------------------------------
__global__ void init_ws_kernel(float* acc, unsigned* colMinBits, int P) {
    int i = threadIdx.x;
    if (i < 4) acc[i] = 0.0f;
    if (i < P) colMinBits[i] = 0x7f800000u; // +inf
}

// ---------------------------------------------------------------------------
// Cross-entropy: one wave per row (C=100 -> 4 strided classes per lane)
// ---------------------------------------------------------------------------
__global__ __launch_bounds__(256) void ce_kernel(const float* __restrict__ preds,
                                                 const int* __restrict__ labels,
                                                 float* __restrict__ ceAcc,
                                                 int B, int C) {
    const int lane = threadIdx.x & 31;
    const int wave = blockIdx.x * (blockDim.x >> 5) + (threadIdx.x >> 5);
    const int nWaves = gridDim.x * (blockDim.x >> 5);
    float sum = 0.0f;
    for (int row = wave; row < B; row += nWaves) {
        const float* p = preds + (size_t)row * C;
        float x[4];
        float m = -__builtin_inff();
        #pragma unroll
        for (int i = 0; i < 4; ++i) {
            int cc = lane + 32 * i;
            x[i] = (cc < C) ? p[cc] : -__builtin_inff();
            m = fmaxf(m, x[i]);
        }
        #pragma unroll
        for (int off = 16; off; off >>= 1) m = fmaxf(m, __shfl_xor(m, off, 32));
        float s = 0.0f;
        #pragma unroll
        for (int i = 0; i < 4; ++i) {
            int cc = lane + 32 * i;
            if (cc < C) s += __expf(x[i] - m);
        }
        s = wave_sum32(s);
        if (lane == 0) {
            int lbl = labels[row];
            sum += (__logf(s) + m - p[lbl]);
        }
    }
    if (lane == 0) atomicAdd(ceAcc, sum);
}

// ---------------------------------------------------------------------------
// MSE: grid-stride float4 streaming (the traffic-dominant kernel, ~256 MB)
// ---------------------------------------------------------------------------
__global__ __launch_bounds__(256) void mse_kernel(const float4* __restrict__ rec,
                                                  const float4* __restrict__ dat,
                                                  float* __restrict__ mseAcc,
                                                  long n4) {
    long i = (long)blockIdx.x * blockDim.x + threadIdx.x;
    const long stride = (long)gridDim.x * blockDim.x;
    float sum = 0.0f;
    for (; i < n4; i += stride) {
        float4 r = rec[i];
        float4 t = dat[i];
        float dx = r.x - t.x, dy = r.y - t.y, dz = r.z - t.z, dw = r.w - t.w;
        sum = fmaf(dx, dx, sum);
        sum = fmaf(dy, dy, sum);
        sum = fmaf(dz, dz, sum);
        sum = fmaf(dw, dw, sum);
    }
    sum = wave_sum32(sum);
    if ((threadIdx.x & 31) == 0) atomicAdd(mseAcc, sum);
}

// ---------------------------------------------------------------------------
// Prototype diversity: one wave per 16x16 tile of the PxP distance matrix,
// GEMM part via V_WMMA_F32_16X16X4_F32 (exact fp32 WMMA). Tiny (P=128).
// ---------------------------------------------------------------------------
__global__ __launch_bounds__(256) void diversity_kernel(const float* __restrict__ Pr,
                                                        float* __restrict__ divAcc,
                                                        int d, int P) {
    const int lane = threadIdx.x & 31;
    const int wave = blockIdx.x * (blockDim.x >> 5) + (threadIdx.x >> 5);
    const int tilesPerRow = P >> 4;
    const int rt = wave / tilesPerRow;
    const int ct = wave % tilesPerRow;
    if (rt >= tilesPerRow) return;
    if (ct < rt) return; // tile entirely below diagonal

    const int r = lane & 15;
    const int hi = lane >> 4;           // lane group
    const int koff = hi * 2;            // K lane offset per ISA f32 A/B layout

    const float* aRow = Pr + (size_t)(rt * 16 + r) * d + koff;
    const float* bRow = Pr + (size_t)(ct * 16 + r) * d + koff;

    v8f acc = (v8f){0, 0, 0, 0, 0, 0, 0, 0};
    float aN = 0.0f, bN = 0.0f;
    for (int k = 0; k < d; k += 4) {
        v2f a = *(const v2f*)(aRow + k);
        v2f b = *(const v2f*)(bRow + k);
        aN = fmaf(a.x, a.x, fmaf(a.y, a.y, aN));
        bN = fmaf(b.x, b.x, fmaf(b.y, b.y, bN));
        acc = __builtin_amdgcn_wmma_f32_16x16x4_f32(false, a, false, b,
                                                    (short)0, acc, false, false);
    }
    float aNf = aN + __shfl_xor(aN, 16, 32); // lane l: ||A row (l&15)||^2
    float bNf = bN + __shfl_xor(bN, 16, 32);

    float sum = 0.0f;
    const int gn = ct * 16 + r;
    #pragma unroll
    for (int j = 0; j < 8; ++j) {
        int mloc = j + (hi << 3);
        float en = __shfl(aNf, mloc, 32);
        int gm = rt * 16 + mloc;
        float sq = en + bNf - 2.0f * acc[j];
        float dist = sqrtf(fmaxf(sq, EPS_D));
        if (gn > gm) {
            float h = fmaxf(0.0f, D_MIN - dist);
            sum = fmaf(h, h, sum);
        }
    }
    sum = wave_sum32(sum);
    if (lane == 0) atomicAdd(divAcc, sum);
}

// ---------------------------------------------------------------------------
// Embedding<->prototype distances.
// Block = 8 waves = 128 embedding rows x all 128 prototypes.
// Prototypes staged in LDS one 64-K slice at a time with CDNA5 async
// global->LDS copies (global_load_async_to_lds_b128 / s_wait_asynccnt);
// GEMM via V_WMMA_F32_16X16X4_F32; row/col mins fused, [B,P] never hits mem.
// ---------------------------------------------------------------------------
__global__ __launch_bounds__(256) void emb_dist_kernel(const float* __restrict__ E,
                                                       const float* __restrict__ Pr,
                                                       float* __restrict__ covAcc,
                                                       unsigned* __restrict__ colMinBits,
                                                       int B, int d) {
    __shared__ float ldsP[NPROTO * ROWPITCH];   // 128 x 66 floats = 33.8 KB

    const int tid  = threadIdx.x;
    const int lane = tid & 31;
    const int rowBase = (blockIdx.x * 8 + (tid >> 5)) * 16;

    const int r = lane & 15;
    const int hi = lane >> 4;
    const int koff = hi * 2;                    // K lane offset (ISA f32 layout)

    const float* eRow = E + (size_t)(rowBase + r) * d + koff;
    const unsigned ldsBase = (unsigned)(uintptr_t)(void*)ldsP;

    v8f acc[8];
    #pragma unroll
    for (int c = 0; c < 8; ++c) acc[c] = (v8f){0, 0, 0, 0, 0, 0, 0, 0};
    float eN = 0.0f;
    float pN[8];
    #pragma unroll
    for (int c = 0; c < 8; ++c) pN[c] = 0.0f;

    const bool active = (rowBase < B);

    for (int kc = 0; kc < d; kc += KC) {
        __syncthreads();  // previous slice fully consumed
        // Cooperative async copy: 128 protos x KC floats = 2048 float4,
        // 8 global_load_async_to_lds_b128 per thread.
        #pragma unroll
        for (int t = 0; t < (NPROTO * KC / 4) / 256; ++t) {
            int idx   = tid + t * 256;          // float4 index in slice
            int proto = idx >> 4;               // KC/4 = 16 float4 per row
            int k4    = idx & 15;
            unsigned ldsOff = ldsBase + (unsigned)(proto * ROWPITCH + k4 * 4) * 4u;
            unsigned long long ga =
                (unsigned long long)(const void*)(Pr + (size_t)proto * d + kc + k4 * 4);
            asm volatile("global_load_async_to_lds_b128 %0, %1, off"
                         :: "v"(ldsOff), "v"(ga) : "memory");
        }
        // prefetch next A-slice while the async DMA runs
        if (kc + KC < d) __builtin_prefetch(eRow + kc + KC, 0, 1);
        asm volatile("s_wait_asynccnt 0x0" ::: "memory");
        __syncthreads();  // slice visible to all waves

        if (active) {
            for (int k = 0; k < KC; k += 4) {
                v2f a = *(const v2f*)(eRow + kc + k);
                eN = fmaf(a.x, a.x, fmaf(a.y, a.y, eN));
                #pragma unroll
                for (int c = 0; c < 8; ++c) {
                    const float* bp = &ldsP[(c * 16 + r) * ROWPITCH + k + koff];
                    v2f b = *(const v2f*)bp;    // ds_load_b64, conflict-free
                    pN[c] = fmaf(b.x, b.x, fmaf(b.y, b.y, pN[c]));
                    acc[c] = __builtin_amdgcn_wmma_f32_16x16x4_f32(
                        false, a, false, b, (short)0, acc[c], false, false);
                }
            }
        }
    }

    if (!active) return;

    float eNf = eN + __shfl_xor(eN, 16, 32); // lane l: ||E[rowBase+(l&15)]||^2
    float pNf[8];
    #pragma unroll
    for (int c = 0; c < 8; ++c) pNf[c] = pN[c] + __shfl_xor(pN[c], 16, 32);

    float covSum = 0.0f;
    float smin[8];
    #pragma unroll
    for (int c = 0; c < 8; ++c) smin[c] = __builtin_inff();

    #pragma unroll
    for (int j = 0; j < 8; ++j) {
        int mloc = j + (hi << 3);
        float en = __shfl(eNf, mloc, 32);
        float rmin = __builtin_inff();
        #pragma unroll
        for (int c = 0; c < 8; ++c) {
            float sq = en + pNf[c] - 2.0f * acc[c][j];
            float dist = sqrtf(fmaxf(sq, EPS_D));
            rmin = fminf(rmin, dist);
            smin[c] = fminf(smin[c], dist);
        }
        rmin = fminf(rmin, __shfl_xor(rmin, 1, 32));
        rmin = fminf(rmin, __shfl_xor(rmin, 2, 32));
        rmin = fminf(rmin, __shfl_xor(rmin, 4, 32));
        rmin = fminf(rmin, __shfl_xor(rmin, 8, 32));
        if ((lane & 15) == 0) covSum += rmin;   // lane0: row j, lane16: row j+8
    }
    if ((lane & 15) == 0) atomicAdd(covAcc, covSum);

    #pragma unroll
    for (int c = 0; c < 8; ++c) {
        float s = fminf(smin[c], __shfl_xor(smin[c], 16, 32));
        if (lane < 16) atomicMin(colMinBits + c * 16 + lane, __float_as_uint(s));
    }
}

// ---------------------------------------------------------------------------
// Final combine (tiny)
// ---------------------------------------------------------------------------
__global__ void final_kernel(const float* __restrict__ acc,
                             const unsigned* __restrict__ colMinBits,
                             float* __restrict__ out, int B, int D, int P) {
    if (threadIdx.x == 0) {
        float simSum = 0.0f;
        for (int i = 0; i < P; ++i) simSum += __uint_as_float(colMinBits[i]);
        float ce  = acc[0] / (float)B;
        float mse = acc[1] / ((float)B * (float)D);
        float div = acc[2];
        float cov = acc[3] / (float)B;
        out[0] = ce + LAM * mse + LAM * div + LAM * (simSum / (float)P) + LAM * cov;
    }
}

extern "C" void kernel_launch(void* const* d_in, const int* in_sizes, int n_in,
                              void* d_out, int out_size, void* d_ws, size_t ws_size,
                              hipStream_t stream) {
    const float* preds  = (const float*)d_in[0];
    const float* rec    = (const float*)d_in[1];
    const float* data   = (const float*)d_in[2];
    const float* emb    = (const float*)d_in[3];
    const float* prot   = (const float*)d_in[4];
    const int*   labels = (const int*)d_in[5];

    const int B = in_sizes[5];              // 32768
    const int C = in_sizes[0] / B;          // 100
    const int D = in_sizes[1] / B;          // 1024
    const int d = in_sizes[3] / B;          // 512
    const int P = in_sizes[4] / d;          // 128

    float*    acc     = (float*)d_ws;
    unsigned* colMin  = (unsigned*)((char*)d_ws + 4 * sizeof(float));

    init_ws_kernel<<<1, 256, 0, stream>>>(acc, colMin, P);

    ce_kernel<<<512, 256, 0, stream>>>(preds, labels, acc + 0, B, C);

    const long n4 = (long)B * D / 4;
    mse_kernel<<<2048, 256, 0, stream>>>((const float4*)rec, (const float4*)data,
                                         acc + 1, n4);

    const int divWaves = (P / 16) * (P / 16);           // 64 waves
    diversity_kernel<<<(divWaves + 7) / 8, 256, 0, stream>>>(prot, acc + 2, d, P);

    const int embBlocks = (B + 127) / 128;              // 8 waves (128 rows) per block
    emb_dist_kernel<<<embBlocks, 256, 0, stream>>>(emb, prot, acc + 3, colMin, B, d);

    final_kernel<<<1, 32, 0, stream>>>(acc, colMin, (float*)d_out, B, D, P);
}